// ModalAttention_40329742909876
// MI455X (gfx1250) — compile-verified
//
#include <hip/hip_runtime.h>
#include <cfloat>
#include <math.h>

typedef float v2f __attribute__((ext_vector_type(2)));
typedef float v8f __attribute__((ext_vector_type(8)));

static constexpr int  kBS  = 32;
static constexpr int  kSeq = 512;
static constexpr int  kHid = 1024;
static constexpr long kOutRow = (long)kSeq * kHid + (long)kSeq * kSeq; // 786432 floats per batch row of d_out
static constexpr int  kLdsW = 36; // padded LDS row stride in floats (even, mult of 4 -> 8B/16B aligned, bank-spread)

__device__ __forceinline__ unsigned encOrd(float f) {
  unsigned u = __float_as_uint(f);
  return (u & 0x80000000u) ? ~u : (u | 0x80000000u); // order-preserving map float -> uint
}
__device__ __forceinline__ float decOrd(unsigned e) {
  unsigned u = (e & 0x80000000u) ? (e ^ 0x80000000u) : ~e;
  return __uint_as_float(u);
}

// ---------------------------------------------------------------------------
// Core 128(M) x 64(N) tile GEMM using V_WMMA_F32_16X16X4_F32.
// 256 threads = 8 waves; wave w computes rows [m0+16w, m0+16w+16) x all 64 cols
// as 4 accumulators of 16x16.
//   B_IS_NT = true : B supplied as W[n][k] (k contiguous)  -> D = A * W^T
//   B_IS_NT = false: B supplied as B[k][n] (n contiguous)  -> D = A * B
// res (optional, same leading dim as C): added before store.
// TRACK_MIN: per-tile min of alpha-scaled result -> atomicMin(encoded).
// ---------------------------------------------------------------------------
template <bool TRACK_MIN, bool B_IS_NT>
__device__ __forceinline__ void gemm_tile_128x64(
    const float* __restrict__ A, int lda,
    const float* __restrict__ B, int ldb,
    const float* __restrict__ res,
    float* __restrict__ C, int ldc,
    int K, float alpha, unsigned* minEnc)
{
  __shared__ float lsA[128 * kLdsW];
  __shared__ float lsB[64 * kLdsW];
  __shared__ float sMin[8];

  const int tid  = threadIdx.x;
  const int wave = tid >> 5;
  const int lane = tid & 31;
  const int lh   = lane & 15;   // lane within half-wave (M for A frag, N for B frag)
  const int hi   = lane >> 4;   // half-wave select (K pair for frags, +8 rows for C/D)

  const int m0 = blockIdx.x * 128;
  const int n0 = blockIdx.y * 64;

  v8f acc[4] = {};

  for (int kc = 0; kc < K; kc += 32) {
    // Stage A: 128 x 32, k-contiguous rows -> lsA[m][k]
#pragma unroll
    for (int i = 0; i < 4; ++i) {
      int idx = tid + i * 256;            // 0..1023 over (row, k-vec4)
      int row = idx >> 3;
      int kv  = (idx & 7) << 2;
      float4 t = *(const float4*)(A + (size_t)(m0 + row) * lda + kc + kv);
      *(float4*)&lsA[row * kLdsW + kv] = t;
    }
    if constexpr (B_IS_NT) {
      // W[n][k], k contiguous: stage directly as lsB[n][k]
#pragma unroll
      for (int i = 0; i < 2; ++i) {
        int idx = tid + i * 256;          // 0..511 over (n, k-vec4)
        int n  = idx >> 3;
        int kv = (idx & 7) << 2;
        float4 t = *(const float4*)(B + (size_t)(n0 + n) * ldb + kc + kv);
        *(float4*)&lsB[n * kLdsW + kv] = t;
      }
    } else {
      // B[k][n], n contiguous: transpose into lsB[n][k]
#pragma unroll
      for (int i = 0; i < 2; ++i) {
        int idx = tid + i * 256;          // 0..511 over (k, n-vec4)
        int kk = idx >> 4;
        int nv = (idx & 15) << 2;
        float4 t = *(const float4*)(B + (size_t)(kc + kk) * ldb + n0 + nv);
        lsB[(nv + 0) * kLdsW + kk] = t.x;
        lsB[(nv + 1) * kLdsW + kk] = t.y;
        lsB[(nv + 2) * kLdsW + kk] = t.z;
        lsB[(nv + 3) * kLdsW + kk] = t.w;
      }
    }
    // Prefetch next K-chunk into cache (global_prefetch_b8)
    if (kc + 32 < K) {
      if (tid < 128) {
        __builtin_prefetch(A + (size_t)(m0 + tid) * lda + kc + 32, 0, 1);
      } else if (tid < 192) {
        int r = tid - 128;
        if constexpr (B_IS_NT) {
          __builtin_prefetch(B + (size_t)(n0 + r) * ldb + kc + 32, 0, 1);
        } else {
          if (r < 32) __builtin_prefetch(B + (size_t)(kc + 32 + r) * ldb + n0, 0, 1);
        }
      }
    }
    __syncthreads();

    // Inner product: 8 WMMA K-steps of 4 per chunk, 4 N-tiles each.
    // f32 fragment layouts (ISA 7.12.2):
    //   A 16x4: lane M = lh, (v0,v1) = K (kb, kb+1) with kb = kk + 2*hi
    //   B 4x16: lane N = lh, (v0,v1) = K (kb, kb+1) with kb = kk + 2*hi
    const float* aBase = &lsA[(wave * 16 + lh) * kLdsW];
    const float* bBase = &lsB[lh * kLdsW];
#pragma unroll
    for (int kk = 0; kk < 32; kk += 4) {
      const int kb = kk + (hi << 1);
      v2f af = *(const v2f*)(aBase + kb);
#pragma unroll
      for (int j = 0; j < 4; ++j) {
        v2f bf = *(const v2f*)(bBase + j * 16 * kLdsW + kb);
        acc[j] = __builtin_amdgcn_wmma_f32_16x16x4_f32(
            false, af, false, bf, (short)0, acc[j], false, false);
      }
    }
    __syncthreads();
  }

#pragma unroll
  for (int j = 0; j < 4; ++j) acc[j] *= alpha;

  // C/D layout: VGPR i -> row (i + 8*hi), col = lh within 16x16 tile
  const int mBase = m0 + wave * 16 + (hi << 3);
#pragma unroll
  for (int j = 0; j < 4; ++j) {
    const int n = n0 + j * 16 + lh;
#pragma unroll
    for (int i = 0; i < 8; ++i) {
      const size_t off = (size_t)(mBase + i) * ldc + n;
      float val = acc[j][i];
      if (res != nullptr) val += res[off];
      C[off] = val;
    }
  }

  if (TRACK_MIN) {
    float lm = FLT_MAX;
#pragma unroll
    for (int j = 0; j < 4; ++j)
#pragma unroll
      for (int i = 0; i < 8; ++i) lm = fminf(lm, acc[j][i]);
#pragma unroll
    for (int o = 16; o > 0; o >>= 1) lm = fminf(lm, __shfl_xor(lm, o));
    if (lane == 0) sMin[wave] = lm;
    __syncthreads();
    if (tid == 0) {
      float m = sMin[0];
#pragma unroll
      for (int i = 1; i < 8; ++i) m = fminf(m, sMin[i]);
      atomicMin(minEnc, encOrd(m));
    }
  }
}

// ---------------------------------------------------------------------------
// Kernels
// ---------------------------------------------------------------------------
__global__ void init_min_kernel(unsigned* minEnc) { *minEnc = 0xFFFFFFFFu; }

__global__ __launch_bounds__(256) void qkv_proj_kernel(
    const float* __restrict__ data,
    const float* __restrict__ wq, const float* __restrict__ wk, const float* __restrict__ wv,
    float* __restrict__ q, float* __restrict__ k, float* __restrict__ v)
{
  const float* W;
  float* C;
  if (blockIdx.z == 0)      { W = wq; C = q; }
  else if (blockIdx.z == 1) { W = wk; C = k; }
  else                      { W = wv; C = v; }
  gemm_tile_128x64<false, true>(data, kHid, W, kHid, nullptr, C, kHid, kHid, 1.0f, nullptr);
}

__global__ __launch_bounds__(256) void attn_scores_kernel(
    const float* __restrict__ q, const float* __restrict__ k,
    float* __restrict__ dout, unsigned* minEnc)
{
  const int b = blockIdx.z;
  const float* A = q + (size_t)b * kSeq * kHid;
  const float* B = k + (size_t)b * kSeq * kHid;                 // scores = (Q/32) * K^T
  float* C = dout + (size_t)b * kOutRow + (size_t)kSeq * kHid;  // raw scores staged in d_out attn region
  gemm_tile_128x64<true, true>(A, kHid, B, kHid, nullptr, C, kSeq, kHid, 0.03125f, minEnc);
}

__device__ __forceinline__ float block_reduce_sum256(float v, float* s) {
#pragma unroll
  for (int o = 16; o > 0; o >>= 1) v += __shfl_xor(v, o);
  const int wave = threadIdx.x >> 5, lane = threadIdx.x & 31;
  if (lane == 0) s[wave] = v;
  __syncthreads();
  float r = s[0];
#pragma unroll
  for (int i = 1; i < 8; ++i) r += s[i];
  __syncthreads();
  return r;
}
__device__ __forceinline__ float block_reduce_max256(float v, float* s) {
#pragma unroll
  for (int o = 16; o > 0; o >>= 1) v = fmaxf(v, __shfl_xor(v, o));
  const int wave = threadIdx.x >> 5, lane = threadIdx.x & 31;
  if (lane == 0) s[wave] = v;
  __syncthreads();
  float r = s[0];
#pragma unroll
  for (int i = 1; i < 8; ++i) r = fmaxf(r, s[i]);
  __syncthreads();
  return r;
}

// In place on d_out attn region: /|globalmin|, L2-normalize (eps 1e-12), softmax.
__global__ __launch_bounds__(256) void attn_softmax_kernel(
    float* __restrict__ dout, const unsigned* __restrict__ minEnc)
{
  __shared__ float sred[8];
  const int row = blockIdx.x;
  const int b = row >> 9, s = row & (kSeq - 1);
  float* x = dout + (size_t)b * kOutRow + (size_t)kSeq * kHid + (size_t)s * kSeq;
  const int tid = threadIdx.x;

  const float c  = fmaxf(fabsf(decOrd(*minEnc)), 1e-30f);
  float t0 = x[tid] / c;
  float t1 = x[tid + 256] / c;

  const float ss = block_reduce_sum256(t0 * t0 + t1 * t1, sred);
  const float dN = fmaxf(sqrtf(ss), 1e-12f);
  const float u0 = t0 / dN, u1 = t1 / dN;

  const float mx  = block_reduce_max256(fmaxf(u0, u1), sred);
  const float e0  = expf(u0 - mx), e1 = expf(u1 - mx);
  const float sum = block_reduce_sum256(e0 + e1, sred);
  const float inv = 1.0f / sum;
  x[tid]       = e0 * inv;
  x[tid + 256] = e1 * inv;
}

__global__ __launch_bounds__(256) void attn_v_kernel(
    const float* __restrict__ dout, const float* __restrict__ v,
    const float* __restrict__ data, float* __restrict__ outPre)
{
  const int b = blockIdx.z;
  const float* A   = dout + (size_t)b * kOutRow + (size_t)kSeq * kHid; // softmaxed attn [512x512]
  const float* B   = v    + (size_t)b * kSeq * kHid;                   // v [512x1024], k-major
  const float* res = data + (size_t)b * kSeq * kHid;                   // residual
  float* C = outPre + (size_t)b * kSeq * kHid;
  gemm_tile_128x64<false, false>(A, kSeq, B, kHid, res, C, kHid, kSeq, 1.0f, nullptr);
}

__global__ __launch_bounds__(256) void layernorm_kernel(
    const float* __restrict__ xin, const float* __restrict__ gamma,
    const float* __restrict__ beta, float* __restrict__ dout)
{
  __shared__ float sred[8];
  const int row = blockIdx.x;
  const int b = row >> 9, s = row & (kSeq - 1);
  const float* x = xin + (size_t)row * kHid;
  float* y = dout + (size_t)b * kOutRow + (size_t)s * kHid;
  const int tid = threadIdx.x;

  const float4 v = *(const float4*)(x + tid * 4);
  const float sum = block_reduce_sum256(v.x + v.y + v.z + v.w, sred);
  const float sq  = block_reduce_sum256(v.x * v.x + v.y * v.y + v.z * v.z + v.w * v.w, sred);
  const float mean = sum * (1.0f / kHid);
  const float var  = sq * (1.0f / kHid) - mean * mean;
  const float rstd = rsqrtf(var + 1e-6f);

  const float4 g  = *(const float4*)(gamma + tid * 4);
  const float4 be = *(const float4*)(beta + tid * 4);
  float4 o;
  o.x = (v.x - mean) * rstd * g.x + be.x;
  o.y = (v.y - mean) * rstd * g.y + be.y;
  o.z = (v.z - mean) * rstd * g.z + be.z;
  o.w = (v.w - mean) * rstd * g.w + be.w;
  *(float4*)(y + tid * 4) = o;
}

// ---------------------------------------------------------------------------
// Host launch. Scratch layout (floats): q[16M] | k[16M] | v[16M] | minEnc(u32)
// = 192 MB + 4 B. outPre reuses q (dead after scores pass). Raw scores live in
// d_out's attn region and are softmaxed in place.
// ---------------------------------------------------------------------------
extern "C" void kernel_launch(void* const* d_in, const int* in_sizes, int n_in,
                              void* d_out, int out_size, void* d_ws, size_t ws_size,
                              hipStream_t stream) {
  (void)in_sizes; (void)n_in; (void)out_size; (void)ws_size;
  const float* data  = (const float*)d_in[0];
  const float* wq    = (const float*)d_in[1];
  const float* wk    = (const float*)d_in[2];
  const float* wv    = (const float*)d_in[3];
  const float* gamma = (const float*)d_in[4];
  const float* beta  = (const float*)d_in[5];
  float* dout = (float*)d_out;

  const size_t tok = (size_t)kBS * kSeq; // 16384
  float* q = (float*)d_ws;
  float* k = q + tok * kHid;
  float* v = k + tok * kHid;
  unsigned* minEnc = (unsigned*)(v + tok * kHid);
  float* outPre = q; // reuse

  init_min_kernel<<<1, 1, 0, stream>>>(minEnc);
  qkv_proj_kernel<<<dim3(128, 16, 3), 256, 0, stream>>>(data, wq, wk, wv, q, k, v);
  attn_scores_kernel<<<dim3(4, 8, kBS), 256, 0, stream>>>(q, k, dout, minEnc);
  attn_softmax_kernel<<<dim3(kBS * kSeq), 256, 0, stream>>>(dout, minEnc);
  attn_v_kernel<<<dim3(4, 16, kBS), 256, 0, stream>>>(dout, v, data, outPre);
  layernorm_kernel<<<dim3(kBS * kSeq), 256, 0, stream>>>(outPre, gamma, beta, dout);
}